// Custom_Element_Wise_48395691491512
// MI455X (gfx1250) — compile-verified
//
#include <hip/hip_runtime.h>
#include <stdint.h>

// Fused int8 dequant -> elementwise multiply -> requant for MI455X (gfx1250).
//
// Shapes (from reference): a,b : [16,4096,1024] int8 (delivered as int32),
// scales : [1024] f32. Output tuple: (out_q int8 -> stored as f32 [64M],
// scale_out f32 [1024]) concatenated flat in d_out.
//
// Memory-bound: ~768 MB of mandatory traffic @ 23.3 TB/s ~= 33 us. The
// CDNA5-specific path exercised here is the async global->LDS DMA
// (ASYNCcnt-tracked), double buffered per block, with per-wave waits only
// (each lane reads back exactly the 16B it asked the DMA to place, so no
// workgroup barriers are needed).

#define TPB 256
#define ROWS_PER_BLOCK 8
#define ROW_ELEMS 1024  // D

// Native clang vector types (required by __builtin_nontemporal_store; HIP's
// float4 is a struct and is rejected).
typedef float v4f __attribute__((ext_vector_type(4)));
typedef int   v4i __attribute__((ext_vector_type(4)));

// Tiny kernel: second tuple output = scale_out passthrough.
__global__ __launch_bounds__(TPB) void scale_tail_kernel(
    const float* __restrict__ so, float* __restrict__ out_tail, int D) {
  int i = blockIdx.x * TPB + threadIdx.x;
  if (i < D) out_tail[i] = so[i];
}

// gfx1250 async DMA: global -> LDS, tracked by ASYNCcnt.
// VDST = per-lane LDS byte address. CRITICAL: lds_addr must be derived from
// the shared variable's address via ptrtoint (capture!), so the "memory"
// clobber is ModRef on the LDS object and later reads can't be folded away.
// (An "m" operand on an LDS address ISel-crashes; an un-escaped integer
// offset lets the optimizer delete the LDS reads entirely.)
__device__ __forceinline__ void async_cp_b128(uint32_t lds_addr, const void* gptr) {
#if defined(__HIP_DEVICE_COMPILE__)
  asm volatile("global_load_async_to_lds_b128 %0, %1, off"
               :: "v"(lds_addr), "v"((uint64_t)(uintptr_t)gptr)
               : "memory");
#endif
}
__device__ __forceinline__ void wait_asynccnt_le2() {
#if defined(__HIP_DEVICE_COMPILE__)
  asm volatile("s_wait_asynccnt 0x2" ::: "memory");
#endif
}
__device__ __forceinline__ void wait_asynccnt_le0() {
#if defined(__HIP_DEVICE_COMPILE__)
  asm volatile("s_wait_asynccnt 0x0" ::: "memory");
#endif
}

// LDS read-back on the DS pipe: explicit addrspace(3) cast -> ds_load_b128
// (a volatile load through the generic pointer lowers to flat_load_b128,
// which burns both LOADcnt and DScnt -- the slow path). volatile because the
// DMA write is invisible to the compiler.
__device__ __forceinline__ v4i lds_read_b128(const v4i* p) {
#if defined(__HIP_DEVICE_COMPILE__)
  return *(__attribute__((address_space(3))) volatile const v4i*)p;
#else
  return *p;
#endif
}

__global__ __launch_bounds__(TPB) void fused_dequant_mul_requant_kernel(
    const int* __restrict__ a, const int* __restrict__ b,
    const float* __restrict__ sa, const float* __restrict__ sb,
    const float* __restrict__ so, float* __restrict__ out) {
  // LDS: 1024 x 16B = 16KB (of 320KB/WGP).
  // Element layout (each chunk is 256 slots = one row's worth, 16B/thread):
  //   [   0..255 ] a-row, buffer 0      [ 256..511 ] b-row, buffer 0
  //   [ 512..767 ] a-row, buffer 1      [ 768..1023] b-row, buffer 1
  __shared__ v4i smem[1024];

  const int tid = threadIdx.x;

  // Thread t owns d = 4t..4t+3 for every row this block touches, so the
  // combined requant scale sa*sb/so lives in registers across all rows.
  // Scales are 12KB total -> L2 resident; 4 divides/thread once per block.
  const v4f a4 = *(const v4f*)(sa + tid * 4);
  const v4f b4 = *(const v4f*)(sb + tid * 4);
  const v4f o4 = *(const v4f*)(so + tid * 4);
  v4f c;
  c.x = a4.x * b4.x / o4.x;
  c.y = a4.y * b4.y / o4.y;
  c.z = a4.z * b4.z / o4.z;
  c.w = a4.w * b4.w / o4.w;

  // This lane's slot in each buffer. The generic address of an LDS object is
  // {aperture_hi32, lds_byte_addr}; truncating to 32 bits gives the DS/async
  // address, and the ptrtoint makes smem escape into the asm (see above).
  v4i* const pa[2] = { &smem[tid],       &smem[512 + tid] };
  v4i* const pb[2] = { &smem[256 + tid], &smem[768 + tid] };

  const long row0 = (long)blockIdx.x * ROWS_PER_BLOCK;

  // Prologue: stage row0 into buffer 0.
  {
    const long e = row0 * ROW_ELEMS + tid * 4;
    async_cp_b128((uint32_t)(uintptr_t)pa[0], a + e);
    async_cp_b128((uint32_t)(uintptr_t)pb[0], b + e);
  }

  for (int r = 0; r < ROWS_PER_BLOCK; ++r) {
    const int cur = r & 1;
    if (r + 1 < ROWS_PER_BLOCK) {
      // Issue next row into the other buffer, then retire current row's pair
      // (async loads complete in order per wave -> <=2 outstanding means the
      // two current-row transfers are done).
      const long e = (row0 + r + 1) * ROW_ELEMS + tid * 4;
      async_cp_b128((uint32_t)(uintptr_t)pa[cur ^ 1], a + e);
      async_cp_b128((uint32_t)(uintptr_t)pb[cur ^ 1], b + e);
      wait_asynccnt_le2();
    } else {
      wait_asynccnt_le0();
    }

    // Each lane reads exactly the 16B it DMA'd -> ds_load_b128, no barrier.
    const v4i av = lds_read_b128(pa[cur]);
    const v4i bv = lds_read_b128(pb[cur]);

    // out = clip(rne(a*b*(sa*sb/so)), -128, 127); rintf == round-half-even.
    v4f o;
    o.x = fminf(fmaxf(rintf((float)av.x * (float)bv.x * c.x), -128.f), 127.f);
    o.y = fminf(fmaxf(rintf((float)av.y * (float)bv.y * c.y), -128.f), 127.f);
    o.z = fminf(fmaxf(rintf((float)av.z * (float)bv.z * c.z), -128.f), 127.f);
    o.w = fminf(fmaxf(rintf((float)av.w * (float)bv.w * c.w), -128.f), 127.f);

    // Streaming store: never re-read, keep it out of L2's way (TH=NT).
    const long e = (row0 + r) * ROW_ELEMS + tid * 4;
    __builtin_nontemporal_store(o, (v4f*)(out + e));
  }
}

extern "C" void kernel_launch(void* const* d_in, const int* in_sizes, int n_in,
                              void* d_out, int out_size, void* d_ws, size_t ws_size,
                              hipStream_t stream) {
  // setup_inputs() order: a, scale_a, b, scale_b, scale_out
  const int*   a  = (const int*)d_in[0];
  const float* sa = (const float*)d_in[1];
  const int*   b  = (const int*)d_in[2];
  const float* sb = (const float*)d_in[3];
  const float* so = (const float*)d_in[4];

  const long n = (long)in_sizes[0];  // 64M elements
  const int  D = in_sizes[1];        // 1024
  const long rows = n / D;           // 65536 (multiple of ROWS_PER_BLOCK)

  float* out = (float*)d_out;

  // Tuple output #2: scale_out passthrough at d_out[n .. n+D)
  scale_tail_kernel<<<(D + TPB - 1) / TPB, TPB, 0, stream>>>(so, out + n, D);

  // Main streaming kernel: 8192 blocks x 256 threads (8 wave32 each).
  fused_dequant_mul_requant_kernel<<<(int)(rows / ROWS_PER_BLOCK), TPB, 0, stream>>>(
      a, b, sa, sb, so, out);
}